// GATConv_59536836657837
// MI455X (gfx1250) — compile-verified
//
#include <hip/hip_runtime.h>
#include <math.h>

// GATConv on gfx1250:
//   h = x @ W           -> WMMA f32 16x16x4 GEMM (writes d_out = h)
//   s_i/s_j per node    -> small dot kernel
//   alpha -> segment max / segment sum over (col, head) via atomics
//   out[n] = h[n] * s[n]/(s[n]+1e-16)   (x_j constant per segment => exact collapse)

#define IN_CHN 128
#define OUT_CHN 32
#define N_HEADS 4
#define NEG_SLOPE 0.2f

typedef __attribute__((ext_vector_type(2))) float v2f;
typedef __attribute__((ext_vector_type(8))) float v8f;

// ---------------- Kernel 1: h = x @ W via v_wmma_f32_16x16x4_f32 ----------------
// Block: 256 threads = 8 waves. Block covers 64 rows x 128 cols.
// Each wave owns one 16-wide N slice and 4 M-tiles (16x16 each), reusing each
// B fragment across 4 WMMAs -> W re-read amortized 4x vs a 16-row tile.
__global__ __launch_bounds__(256) void gat_gemm_wmma(const float* __restrict__ x,
                                                     const float* __restrict__ w,
                                                     float* __restrict__ h,
                                                     int nNodes) {
  __shared__ float sA[64][132];   // 64x128 A tile, row stride 132 (conflict-free, 16B aligned)
  const int t = threadIdx.x;
  const int rowBase = blockIdx.x * 64;

  // Stage A tile: 8192 floats, 32 per thread (8x float4), coalesced.
  {
    int r = t >> 2;              // 0..63
    int c = (t & 3) * 32;        // 0,32,64,96
    int rr = rowBase + r;
    if (rr >= nNodes) rr = nNodes - 1;     // clamp (keeps EXEC full before WMMA)
    const float4* src = (const float4*)(x + (size_t)rr * IN_CHN + c);
    #pragma unroll
    for (int q = 0; q < 8; ++q)
      *(float4*)&sA[r][c + 4 * q] = src[q];
  }
  __syncthreads();

  const int lane = t & 31;       // wave32
  const int wv   = t >> 5;       // 8 waves -> 8 N-tiles of 16
  const int mn   = lane & 15;    // M index for A, N index for B/C/D
  const int g    = lane >> 4;    // K-half selector
  const int colBase = wv * 16;

  v8f acc0 = {}, acc1 = {}, acc2 = {}, acc3 = {};
  for (int k0 = 0; k0 < IN_CHN; k0 += 4) {
    const int kk = k0 + 2 * g;
    // B fragment: lane (g,mn) holds W[kk][colBase+mn], W[kk+1][colBase+mn]
    v2f b = { w[(size_t)kk * (N_HEADS * OUT_CHN) + colBase + mn],
              w[(size_t)(kk + 1) * (N_HEADS * OUT_CHN) + colBase + mn] };
    // A fragments: lane (g,mn) holds A[m][kk], A[m][kk+1] for 4 M-tiles
    v2f a0 = { sA[mn][kk],      sA[mn][kk + 1] };
    v2f a1 = { sA[16 + mn][kk], sA[16 + mn][kk + 1] };
    v2f a2 = { sA[32 + mn][kk], sA[32 + mn][kk + 1] };
    v2f a3 = { sA[48 + mn][kk], sA[48 + mn][kk + 1] };
    acc0 = __builtin_amdgcn_wmma_f32_16x16x4_f32(false, a0, false, b, (short)0, acc0, false, false);
    acc1 = __builtin_amdgcn_wmma_f32_16x16x4_f32(false, a1, false, b, (short)0, acc1, false, false);
    acc2 = __builtin_amdgcn_wmma_f32_16x16x4_f32(false, a2, false, b, (short)0, acc2, false, false);
    acc3 = __builtin_amdgcn_wmma_f32_16x16x4_f32(false, a3, false, b, (short)0, acc3, false, false);
  }

  // C/D layout: VGPR v, lanes 0-15 -> M=v, lanes 16-31 -> M=v+8; N = mn.
  const int NROW = N_HEADS * OUT_CHN;   // 128 floats per row
  if (rowBase + 64 <= nNodes) {
    // Fast path (all blocks but the last): unguarded stores, constant offsets.
    float* p = h + (size_t)(rowBase + 8 * g) * NROW + colBase + mn;
    #pragma unroll
    for (int v = 0; v < 8; ++v) {
      p[(size_t)v * NROW]         = acc0[v];
      p[(size_t)(v + 16) * NROW]  = acc1[v];
      p[(size_t)(v + 32) * NROW]  = acc2[v];
      p[(size_t)(v + 48) * NROW]  = acc3[v];
    }
  } else {
    #pragma unroll
    for (int v = 0; v < 8; ++v) {
      int rr = rowBase + v + 8 * g;
      if (rr < nNodes)      h[(size_t)rr * NROW + colBase + mn]        = acc0[v];
      if (rr + 16 < nNodes) h[(size_t)(rr + 16) * NROW + colBase + mn] = acc1[v];
      if (rr + 32 < nNodes) h[(size_t)(rr + 32) * NROW + colBase + mn] = acc2[v];
      if (rr + 48 < nNodes) h[(size_t)(rr + 48) * NROW + colBase + mn] = acc3[v];
    }
  }
}

// ---------------- Kernel 2: per-node attention scores ----------------
// s_i[n,hd] = dot(h[n, hd*32 .. +31], att[hd, 0..31])
// s_j[n,hd] = dot(h[n, hd*32 .. +31], att[hd, 32..63])
__global__ __launch_bounds__(256) void gat_scores(const float* __restrict__ h,
                                                  const float* __restrict__ att,
                                                  float* __restrict__ si,
                                                  float* __restrict__ sj,
                                                  int nNodes) {
  int idx = blockIdx.x * blockDim.x + threadIdx.x;
  if (idx >= nNodes * N_HEADS) return;
  int n  = idx >> 2;
  int hd = idx & 3;
  const float* hp = h + (size_t)n * IN_CHN + hd * OUT_CHN;
  const float* ai = att + (size_t)hd * (2 * OUT_CHN);
  const float* aj = ai + OUT_CHN;
  float acci = 0.0f, accj = 0.0f;
  #pragma unroll
  for (int c = 0; c < OUT_CHN; c += 4) {
    float4 hv = *(const float4*)(hp + c);
    float4 vi = *(const float4*)(ai + c);
    float4 vj = *(const float4*)(aj + c);
    acci += hv.x * vi.x + hv.y * vi.y + hv.z * vi.z + hv.w * vi.w;
    accj += hv.x * vj.x + hv.y * vj.y + hv.z * vj.z + hv.w * vj.w;
  }
  si[idx] = acci;
  sj[idx] = accj;
}

// ---------------- Kernel 3: init segment tables ----------------
__global__ void gat_init(float* __restrict__ amax, float* __restrict__ ssum, int n) {
  int i = blockIdx.x * blockDim.x + threadIdx.x;
  if (i < n) {
    amax[i] = -INFINITY;
    ssum[i] = 0.0f;
  }
}

// Order-preserving float atomic max (sign-split int/uint trick; init must be -inf).
__device__ __forceinline__ void atomicMaxF(float* addr, float v) {
  if (v >= 0.0f) {
    atomicMax((int*)addr, __float_as_int(v));
  } else {
    atomicMin((unsigned int*)addr, __float_as_uint(v));
  }
}

// ---------------- Kernel 4: edge pass A — segment max of leaky-relu(score) ----------------
__global__ __launch_bounds__(256) void gat_edge_max(const long long* __restrict__ ei,
                                                    const float* __restrict__ si,
                                                    const float* __restrict__ sj,
                                                    float* __restrict__ amax,
                                                    int nE) {
  int e = blockIdx.x * blockDim.x + threadIdx.x;
  if (e >= nE) return;
  long long r = ei[e];
  long long c = ei[(size_t)nE + e];
  float4 vi = *(const float4*)(si + r * N_HEADS);
  float4 vj = *(const float4*)(sj + c * N_HEADS);
  float a0 = vi.x + vj.x, a1 = vi.y + vj.y, a2 = vi.z + vj.z, a3 = vi.w + vj.w;
  a0 = (a0 >= 0.0f) ? a0 : NEG_SLOPE * a0;
  a1 = (a1 >= 0.0f) ? a1 : NEG_SLOPE * a1;
  a2 = (a2 >= 0.0f) ? a2 : NEG_SLOPE * a2;
  a3 = (a3 >= 0.0f) ? a3 : NEG_SLOPE * a3;
  float* base = amax + c * N_HEADS;
  atomicMaxF(base + 0, a0);
  atomicMaxF(base + 1, a1);
  atomicMaxF(base + 2, a2);
  atomicMaxF(base + 3, a3);
}

// ---------------- Kernel 5: edge pass B — segment sum of exp(alpha - amax) ----------------
__global__ __launch_bounds__(256) void gat_edge_sum(const long long* __restrict__ ei,
                                                    const float* __restrict__ si,
                                                    const float* __restrict__ sj,
                                                    const float* __restrict__ amax,
                                                    float* __restrict__ ssum,
                                                    int nE) {
  int e = blockIdx.x * blockDim.x + threadIdx.x;
  if (e >= nE) return;
  long long r = ei[e];
  long long c = ei[(size_t)nE + e];
  float4 vi = *(const float4*)(si + r * N_HEADS);
  float4 vj = *(const float4*)(sj + c * N_HEADS);
  float4 mx = *(const float4*)(amax + c * N_HEADS);
  float a0 = vi.x + vj.x, a1 = vi.y + vj.y, a2 = vi.z + vj.z, a3 = vi.w + vj.w;
  a0 = (a0 >= 0.0f) ? a0 : NEG_SLOPE * a0;
  a1 = (a1 >= 0.0f) ? a1 : NEG_SLOPE * a1;
  a2 = (a2 >= 0.0f) ? a2 : NEG_SLOPE * a2;
  a3 = (a3 >= 0.0f) ? a3 : NEG_SLOPE * a3;
  float* base = ssum + c * N_HEADS;
  atomicAdd(base + 0, expf(a0 - mx.x));
  atomicAdd(base + 1, expf(a1 - mx.y));
  atomicAdd(base + 2, expf(a2 - mx.z));
  atomicAdd(base + 3, expf(a3 - mx.w));
}

// ---------------- Kernel 6: out[n] = h[n] * s/(s+1e-16), in place on d_out ----------------
__global__ __launch_bounds__(256) void gat_scale(float* __restrict__ out,
                                                 const float* __restrict__ ssum,
                                                 int total) {
  int i = blockIdx.x * blockDim.x + threadIdx.x;
  if (i >= total) return;
  int n  = i >> 7;          // /128
  int ch = i & 127;
  int hd = ch >> 5;         // /32
  float s = ssum[n * N_HEADS + hd];
  out[i] *= s / (s + 1e-16f);
}

extern "C" void kernel_launch(void* const* d_in, const int* in_sizes, int n_in,
                              void* d_out, int out_size, void* d_ws, size_t ws_size,
                              hipStream_t stream) {
  const float*     x   = (const float*)d_in[0];
  const float*     w   = (const float*)d_in[1];
  const float*     att = (const float*)d_in[2];
  const long long* ei  = (const long long*)d_in[3];

  const int nNodes = in_sizes[0] / IN_CHN;
  const int nE     = in_sizes[3] / 2;
  const int nSeg   = nNodes * N_HEADS;

  float* h = (float*)d_out;                 // d_out doubles as h buffer

  float* si   = (float*)d_ws;               // nSeg floats
  float* sj   = si + nSeg;                  // nSeg floats
  float* amax = sj + nSeg;                  // nSeg floats
  float* ssum = amax + nSeg;                // nSeg floats
  (void)ws_size;

  // 1) GEMM: h = x @ W  (WMMA f32 16x16x4, 64-row M tiles)
  {
    int grid = (nNodes + 63) / 64;
    gat_gemm_wmma<<<grid, 256, 0, stream>>>(x, w, h, nNodes);
  }
  // 2) per-node scores
  {
    int grid = (nSeg + 255) / 256;
    gat_scores<<<grid, 256, 0, stream>>>(h, att, si, sj, nNodes);
  }
  // 3) init amax/-inf, ssum/0
  {
    int grid = (nSeg + 255) / 256;
    gat_init<<<grid, 256, 0, stream>>>(amax, ssum, nSeg);
  }
  // 4) segment max
  {
    int grid = (nE + 255) / 256;
    gat_edge_max<<<grid, 256, 0, stream>>>(ei, si, sj, amax, nE);
  }
  // 5) segment sum of exp
  {
    int grid = (nE + 255) / 256;
    gat_edge_sum<<<grid, 256, 0, stream>>>(ei, si, sj, amax, ssum, nE);
  }
  // 6) scale h -> out in place
  {
    int total = nNodes * IN_CHN;
    int grid = (total + 255) / 256;
    gat_scale<<<grid, 256, 0, stream>>>(h, ssum, total);
  }
}